// GCN_81638738363153
// MI455X (gfx1250) — compile-verified
//
#include <hip/hip_runtime.h>

typedef __attribute__((ext_vector_type(2))) float v2f;
typedef __attribute__((ext_vector_type(8))) float v8f;

#define IN_F  512
#define HID_F 64
#define OUT_F 7

// ---------------- utility kernels ----------------

__global__ void zero_f32(float* __restrict__ p, int n) {
    int i = blockIdx.x * blockDim.x + threadIdx.x;
    if (i < n) p[i] = 0.0f;
}

// deg[dst[e]] += 1 for all E edges plus self loops
__global__ void degree_kernel(const long long* __restrict__ dst, float* __restrict__ deg,
                              int E, int N) {
    int e = blockIdx.x * blockDim.x + threadIdx.x;
    int total = E + N;
    if (e >= total) return;
    int d = (e < E) ? (int)dst[e] : (e - E);
    atomicAdd(&deg[d], 1.0f);
}

__global__ void dinv_kernel(const float* __restrict__ deg, float* __restrict__ dinv, int N) {
    int i = blockIdx.x * blockDim.x + threadIdx.x;
    if (i >= N) return;
    float d = deg[i];
    dinv[i] = (d > 0.0f) ? rsqrtf(fmaxf(d, 1.0f)) : 0.0f;
}

__global__ void norm_kernel(const long long* __restrict__ src, const long long* __restrict__ dst,
                            const float* __restrict__ dinv, float* __restrict__ norm,
                            int E, int N) {
    int e = blockIdx.x * blockDim.x + threadIdx.x;
    int total = E + N;
    if (e >= total) return;
    int s = (e < E) ? (int)src[e] : (e - E);
    int d = (e < E) ? (int)dst[e] : (e - E);
    norm[e] = dinv[s] * dinv[d];
}

// ---------------- GEMM1: h[N,64] = x[N,512] @ W1[512,64]  (f32 WMMA) ----------------
// One wave computes TWO 16-row strips (register blocking over M) so every
// B fragment load feeds two v_wmma ops. 4 waves per 128-thread block.
__global__ __launch_bounds__(128) void gemm1_wmma(const float* __restrict__ x,
                                                  const float* __restrict__ W1,
                                                  float* __restrict__ h, int N) {
    const int wave = threadIdx.x >> 5;
    const int lane = threadIdx.x & 31;
    const int pt   = blockIdx.x * 4 + wave;   // pair-of-tiles index
    const int PT   = (N + 31) / 32;
    if (pt >= PT) return;                     // wave-uniform: EXEC stays full below

    const int l16  = lane & 15;
    const int half = lane >> 4;               // 0: K={0,1}, 1: K={2,3}
    const int mt0  = pt * 2;
    const int mt1  = pt * 2 + 1;
    const int row0 = mt0 * 16 + l16;
    const int row1 = mt1 * 16 + l16;
    const bool ok0 = row0 < N;
    const bool ok1 = row1 < N;
    const int rs0  = ok0 ? row0 : 0;
    const int rs1  = ok1 ? row1 : 0;

    v8f accA0 = {}, accA1 = {}, accA2 = {}, accA3 = {};
    v8f accB0 = {}, accB1 = {}, accB2 = {}, accB3 = {};

    for (int k0 = 0; k0 < IN_F; k0 += 4) {
        const int ka = k0 + 2 * half;
        // A fragments: 16x4 f32; lane holds x[row][ka..ka+1]
        v2f a0 = *(const v2f*)(x + (size_t)rs0 * IN_F + ka);
        v2f a1 = *(const v2f*)(x + (size_t)rs1 * IN_F + ka);
        if (!ok0) { a0.x = 0.0f; a0.y = 0.0f; }
        if (!ok1) { a1.x = 0.0f; a1.y = 0.0f; }
        // B fragments: 4x16 f32; lane holds W1[ka][col], W1[ka+1][col]
        {
            int col = 0 * 16 + l16;
            v2f b; b.x = W1[(size_t)ka * HID_F + col]; b.y = W1[(size_t)(ka + 1) * HID_F + col];
            accA0 = __builtin_amdgcn_wmma_f32_16x16x4_f32(false, a0, false, b, (short)0, accA0, false, false);
            accB0 = __builtin_amdgcn_wmma_f32_16x16x4_f32(false, a1, false, b, (short)0, accB0, false, false);
        }
        {
            int col = 1 * 16 + l16;
            v2f b; b.x = W1[(size_t)ka * HID_F + col]; b.y = W1[(size_t)(ka + 1) * HID_F + col];
            accA1 = __builtin_amdgcn_wmma_f32_16x16x4_f32(false, a0, false, b, (short)0, accA1, false, false);
            accB1 = __builtin_amdgcn_wmma_f32_16x16x4_f32(false, a1, false, b, (short)0, accB1, false, false);
        }
        {
            int col = 2 * 16 + l16;
            v2f b; b.x = W1[(size_t)ka * HID_F + col]; b.y = W1[(size_t)(ka + 1) * HID_F + col];
            accA2 = __builtin_amdgcn_wmma_f32_16x16x4_f32(false, a0, false, b, (short)0, accA2, false, false);
            accB2 = __builtin_amdgcn_wmma_f32_16x16x4_f32(false, a1, false, b, (short)0, accB2, false, false);
        }
        {
            int col = 3 * 16 + l16;
            v2f b; b.x = W1[(size_t)ka * HID_F + col]; b.y = W1[(size_t)(ka + 1) * HID_F + col];
            accA3 = __builtin_amdgcn_wmma_f32_16x16x4_f32(false, a0, false, b, (short)0, accA3, false, false);
            accB3 = __builtin_amdgcn_wmma_f32_16x16x4_f32(false, a1, false, b, (short)0, accB3, false, false);
        }
    }

    // D layout: VGPR r, lanes 0-15 -> M=r, lanes 16-31 -> M=8+r; N = lane&15
    #pragma unroll
    for (int r = 0; r < 8; ++r) {
        int mA = mt0 * 16 + r + 8 * half;
        if (mA < N) {
            h[(size_t)mA * HID_F +  0 + l16] = accA0[r];
            h[(size_t)mA * HID_F + 16 + l16] = accA1[r];
            h[(size_t)mA * HID_F + 32 + l16] = accA2[r];
            h[(size_t)mA * HID_F + 48 + l16] = accA3[r];
        }
        int mB = mt1 * 16 + r + 8 * half;
        if (mB < N) {
            h[(size_t)mB * HID_F +  0 + l16] = accB0[r];
            h[(size_t)mB * HID_F + 16 + l16] = accB1[r];
            h[(size_t)mB * HID_F + 32 + l16] = accB2[r];
            h[(size_t)mB * HID_F + 48 + l16] = accB3[r];
        }
    }
}

// ---------------- scatter layer 1: agg[dst] += h[src] * norm ----------------
// 16 lanes per edge; each lane moves a float4 (row = 64 floats = 16 * float4)
__global__ void scatter1_kernel(const long long* __restrict__ src,
                                const long long* __restrict__ dst,
                                const float* __restrict__ norm,
                                const float* __restrict__ h,
                                float* __restrict__ agg, int E, int N) {
    int t = blockIdx.x * blockDim.x + threadIdx.x;
    int e = t >> 4;
    int c = t & 15;
    if (e >= E + N) return;
    int s = (e < E) ? (int)src[e] : (e - E);
    int d = (e < E) ? (int)dst[e] : (e - E);
    float w = norm[e];
    float4 v = *(const float4*)(h + (size_t)s * HID_F + c * 4);
    float* o = agg + (size_t)d * HID_F + c * 4;
    atomicAdd(o + 0, v.x * w);
    atomicAdd(o + 1, v.y * w);
    atomicAdd(o + 2, v.z * w);
    atomicAdd(o + 3, v.w * w);
}

// ---------------- GEMM2: h2[N,8pad] = relu(agg + b1) @ W2[64,7]  (f32 WMMA) ----------------
__global__ __launch_bounds__(32) void gemm2_wmma(const float* __restrict__ agg,
                                                 const float* __restrict__ b1,
                                                 const float* __restrict__ W2,
                                                 float* __restrict__ h2, int N) {
    const int mt   = blockIdx.x;
    const int lane = threadIdx.x;
    const int l16  = lane & 15;
    const int half = lane >> 4;
    const int row  = mt * 16 + l16;
    const bool rowok = row < N;
    const int rsafe  = rowok ? row : 0;

    v8f acc = {};
    for (int k0 = 0; k0 < HID_F; k0 += 4) {
        const int ka = k0 + 2 * half;
        v2f a = *(const v2f*)(agg + (size_t)rsafe * HID_F + ka);
        a.x = fmaxf(a.x + b1[ka], 0.0f);
        a.y = fmaxf(a.y + b1[ka + 1], 0.0f);
        if (!rowok) { a.x = 0.0f; a.y = 0.0f; }
        v2f b;
        b.x = (l16 < OUT_F) ? W2[(size_t)ka * OUT_F + l16] : 0.0f;
        b.y = (l16 < OUT_F) ? W2[(size_t)(ka + 1) * OUT_F + l16] : 0.0f;
        acc = __builtin_amdgcn_wmma_f32_16x16x4_f32(false, a, false, b, (short)0, acc, false, false);
    }
    #pragma unroll
    for (int r = 0; r < 8; ++r) {
        int m = mt * 16 + r + 8 * half;
        if (m < N && l16 < 8) h2[(size_t)m * 8 + l16] = (l16 < OUT_F) ? acc[r] : 0.0f;
    }
}

// ---------------- init out with bias b2 ----------------
__global__ void init_out_kernel(float* __restrict__ out, const float* __restrict__ b2, int N) {
    int i = blockIdx.x * blockDim.x + threadIdx.x;
    if (i < N * OUT_F) out[i] = b2[i % OUT_F];
}

// ---------------- scatter layer 2: out[dst] += h2[src] * norm ----------------
// 8 lanes per edge (lane 7 idle)
__global__ void scatter2_kernel(const long long* __restrict__ src,
                                const long long* __restrict__ dst,
                                const float* __restrict__ norm,
                                const float* __restrict__ h2,
                                float* __restrict__ out, int E, int N) {
    int t = blockIdx.x * blockDim.x + threadIdx.x;
    int e = t >> 3;
    int c = t & 7;
    if (e >= E + N || c >= OUT_F) return;
    int s = (e < E) ? (int)src[e] : (e - E);
    int d = (e < E) ? (int)dst[e] : (e - E);
    atomicAdd(&out[(size_t)d * OUT_F + c], h2[(size_t)s * 8 + c] * norm[e]);
}

// ---------------- launch ----------------

extern "C" void kernel_launch(void* const* d_in, const int* in_sizes, int n_in,
                              void* d_out, int out_size, void* d_ws, size_t ws_size,
                              hipStream_t stream) {
    const float*     x  = (const float*)d_in[0];
    const long long* ei = (const long long*)d_in[1];
    const float*     W1 = (const float*)d_in[2];
    const float*     b1 = (const float*)d_in[3];
    const float*     W2 = (const float*)d_in[4];
    const float*     b2 = (const float*)d_in[5];
    float* out = (float*)d_out;

    const int N = in_sizes[0] / IN_F;        // 50000
    const int E = in_sizes[1] / 2;           // 1600000
    const int T = E + N;

    const long long* src = ei;
    const long long* dst = ei + E;

    // workspace layout (floats)
    float* ws   = (float*)d_ws;
    float* deg  = ws;                         // N
    float* dinv = deg  + N;                   // N
    float* nrm  = dinv + N;                   // E+N
    float* h    = nrm  + T;                   // N*64
    float* agg  = h    + (size_t)N * HID_F;   // N*64
    float* h2   = agg  + (size_t)N * HID_F;   // N*8
    (void)ws_size; (void)n_in; (void)out_size;

    const int B = 256;

    // graph normalization
    zero_f32<<<(N + B - 1) / B, B, 0, stream>>>(deg, N);
    zero_f32<<<((int)((size_t)N * HID_F) + B - 1) / B, B, 0, stream>>>(agg, N * HID_F);
    degree_kernel<<<(T + B - 1) / B, B, 0, stream>>>(dst, deg, E, N);
    dinv_kernel<<<(N + B - 1) / B, B, 0, stream>>>(deg, dinv, N);
    norm_kernel<<<(T + B - 1) / B, B, 0, stream>>>(src, dst, dinv, nrm, E, N);

    // layer 1
    const int ptiles = (N + 31) / 32;                 // pair-of-16-row-tiles
    gemm1_wmma<<<(ptiles + 3) / 4, 128, 0, stream>>>(x, W1, h, N);
    {
        long long nt = (long long)T * 16;
        scatter1_kernel<<<(int)((nt + B - 1) / B), B, 0, stream>>>(src, dst, nrm, h, agg, E, N);
    }

    // layer 2 (bias+relu fused into GEMM2 A-load)
    const int mtiles = (N + 15) / 16;
    gemm2_wmma<<<mtiles, 32, 0, stream>>>(agg, b1, W2, h2, N);
    init_out_kernel<<<(N * OUT_F + B - 1) / B, B, 0, stream>>>(out, b2, N);
    {
        long long nt = (long long)T * 8;
        scatter2_kernel<<<(int)((nt + B - 1) / B), B, 0, stream>>>(src, dst, nrm, h2, out, E, N);
    }
}